// PScanTriton_49125835931950
// MI455X (gfx1250) — compile-verified
//
#include <hip/hip_runtime.h>
#include <stdint.h>

typedef __attribute__((ext_vector_type(2))) float f32x2;
typedef __attribute__((ext_vector_type(4))) float f32x4;

#define B_      8
#define L_      4096
#define D_      256
#define D2_     (D_ / 2)          // f32x2 elements per row
#define CHUNKS  128
#define LC      (L_ / CHUNKS)     // 32 timesteps per chunk
#define STAGES  8                 // async global->LDS pipeline depth (pass 3)
#define SUBS    8                 // chunk sub-groups per channel (pass 2)
#define CPS     (CHUNKS / SUBS)   // 16 chunks per sub-group

static_assert(LC == 32, "chunking assumption");
static_assert(STAGES <= LC, "pipeline deeper than chunk");

// ---------------------------------------------------------------------------
// Pass 1: per-chunk aggregate (a_prod, x_agg) such that y_out = a*y_in + x.
// 128 threads/block, 2 channels per thread (b64 loads, 256B/wave/array).
// Reads all of A/X once (warms L2 for pass 3), writes 4 MB of aggregates.
// ---------------------------------------------------------------------------
__global__ __launch_bounds__(128) void pscan_chunk_agg(
    const float* __restrict__ A_re, const float* __restrict__ A_im,
    const float* __restrict__ X_re, const float* __restrict__ X_im,
    f32x4* __restrict__ agg)
{
  const int lane = threadIdx.x;             // owns channels d0=2*lane, d0+1
  const int c = blockIdx.x % CHUNKS;
  const int b = blockIdx.x / CHUNKS;
  const size_t row0 = (size_t)b * L_ + (size_t)c * LC;  // first timestep row

  const f32x2* pAr = (const f32x2*)A_re + row0 * D2_ + lane;
  const f32x2* pAi = (const f32x2*)A_im + row0 * D2_ + lane;
  const f32x2* pXr = (const f32x2*)X_re + row0 * D2_ + lane;
  const f32x2* pXi = (const f32x2*)X_im + row0 * D2_ + lane;

  float ar0 = 1.f, ai0 = 0.f, xr0 = 0.f, xi0 = 0.f;   // combine identity
  float ar1 = 1.f, ai1 = 0.f, xr1 = 0.f, xi1 = 0.f;

  for (int to = 0; to < LC; to += 8) {
#pragma unroll
    for (int tt = 0; tt < 8; ++tt) {
      const int i = (to + tt) * D2_;        // folds into instruction offset
      const f32x2 va_r = pAr[i], va_i = pAi[i];
      const f32x2 vx_r = pXr[i], vx_i = pXi[i];
      // combine(earlier=(a,x), later=(ca,cx)) = (ca*a, ca*x + cx)
      {
        const float car = va_r.x, cai = va_i.x, cxr = vx_r.x, cxi = vx_i.x;
        const float nar = fmaf(car, ar0, -cai * ai0);
        const float nai = fmaf(car, ai0,  cai * ar0);
        const float nxr = fmaf(car, xr0, fmaf(-cai, xi0, cxr));
        const float nxi = fmaf(car, xi0, fmaf( cai, xr0, cxi));
        ar0 = nar; ai0 = nai; xr0 = nxr; xi0 = nxi;
      }
      {
        const float car = va_r.y, cai = va_i.y, cxr = vx_r.y, cxi = vx_i.y;
        const float nar = fmaf(car, ar1, -cai * ai1);
        const float nai = fmaf(car, ai1,  cai * ar1);
        const float nxr = fmaf(car, xr1, fmaf(-cai, xi1, cxr));
        const float nxi = fmaf(car, xi1, fmaf( cai, xr1, cxi));
        ar1 = nar; ai1 = nai; xr1 = nxr; xi1 = nxi;
      }
    }
  }
  const size_t o = ((size_t)b * CHUNKS + c) * D_ + 2 * lane;
  f32x4 v0; v0.x = ar0; v0.y = ai0; v0.z = xr0; v0.w = xi0;
  f32x4 v1; v1.x = ar1; v1.y = ai1; v1.z = xr1; v1.w = xi1;
  agg[o]     = v0;
  agg[o + 1] = v1;
}

// ---------------------------------------------------------------------------
// Pass 2: scan the 128 chunk aggregates per channel -> carry-in y per chunk.
// 8-way parallel over chunk sub-groups, LDS exchange of sub-group aggregates.
// ---------------------------------------------------------------------------
__global__ __launch_bounds__(256) void pscan_chunk_carry(
    const f32x4* __restrict__ agg, f32x2* __restrict__ carry)
{
  __shared__ float sA[SUBS][32][4];
  const int ld    = threadIdx.x & 31;   // d within 32-wide tile
  const int sub   = threadIdx.x >> 5;   // 0..7 chunk sub-group
  const int tiles = D_ / 32;            // 8
  const int b  = blockIdx.x / tiles;
  const int d  = (blockIdx.x % tiles) * 32 + ld;
  const int c0 = sub * CPS;

  // Preload my 16 aggregates into registers (independent loads, L2-resident).
  f32x4 r[CPS];
#pragma unroll
  for (int i = 0; i < CPS; ++i)
    r[i] = agg[((size_t)b * CHUNKS + (c0 + i)) * D_ + d];

  // Fold them into this sub-group's aggregate.
  float ar = 1.f, ai = 0.f, xr = 0.f, xi = 0.f;
#pragma unroll
  for (int i = 0; i < CPS; ++i) {
    const float car = r[i].x, cai = r[i].y, cxr = r[i].z, cxi = r[i].w;
    const float nar = fmaf(car, ar, -cai * ai);
    const float nai = fmaf(car, ai,  cai * ar);
    const float nxr = fmaf(car, xr, fmaf(-cai, xi, cxr));
    const float nxi = fmaf(car, xi, fmaf( cai, xr, cxi));
    ar = nar; ai = nai; xr = nxr; xi = nxi;
  }
  sA[sub][ld][0] = ar; sA[sub][ld][1] = ai;
  sA[sub][ld][2] = xr; sA[sub][ld][3] = xi;
  __syncthreads();

  // Exclusive fold over earlier sub-groups: y at my first chunk (y starts 0).
  float yr = 0.f, yi = 0.f;
  for (int s = 0; s < sub; ++s) {
    const float car = sA[s][ld][0], cai = sA[s][ld][1];
    const float cxr = sA[s][ld][2], cxi = sA[s][ld][3];
    const float nyr = fmaf(car, yr, fmaf(-cai, yi, cxr));
    const float nyi = fmaf(car, yi, fmaf( cai, yr, cxi));
    yr = nyr; yi = nyi;
  }

  // Walk my chunks, emitting the carry-in for each.
#pragma unroll
  for (int i = 0; i < CPS; ++i) {
    f32x2 cy; cy.x = yr; cy.y = yi;
    carry[((size_t)b * CHUNKS + (c0 + i)) * D_ + d] = cy;
    const float car = r[i].x, cai = r[i].y, cxr = r[i].z, cxi = r[i].w;
    const float nyr = fmaf(car, yr, fmaf(-cai, yi, cxr));
    const float nyi = fmaf(car, yi, fmaf( cai, yr, cxi));
    yr = nyr; yi = nyi;
  }
}

// ---------------------------------------------------------------------------
// Pass 3: apply recurrence per chunk from carry-in, streaming A/X through an
// 8-stage async pipeline (GLOBAL_LOAD_ASYNC_TO_LDS_B64, th:TH_LOAD_LU since
// this is the last read of L2-resident input; throttled by s_wait_asynccnt).
// LDS layout [stage][array][D] floats; each of 128 threads owns 2 channels.
// Each wave consumes only LDS it wrote -> no barriers needed. The timestep
// loop is fully unrolled so every wait immediate, LDS stage offset and tail
// predicate constant-folds.
// ---------------------------------------------------------------------------
__global__ __launch_bounds__(128) void pscan_apply(
    const float* __restrict__ A_re, const float* __restrict__ A_im,
    const float* __restrict__ X_re, const float* __restrict__ X_im,
    const f32x2* __restrict__ carry,
    f32x4* __restrict__ Y2)               // 2 packed complex per element
{
  __shared__ __align__(16) float sbuf[STAGES][4][D_];   // 32 KB

  const int lane = threadIdx.x;           // channels d0=2*lane, d0+1
  const int d0   = 2 * lane;
  const int c = blockIdx.x % CHUNKS;
  const int b = blockIdx.x / CHUNKS;
  const size_t base = ((size_t)b * L_ + (size_t)c * LC) * D_;

  const float* pAr = A_re + base;
  const float* pAi = A_im + base;
  const float* pXr = X_re + base;
  const float* pXi = X_im + base;

  // LDS byte offset of sbuf: low 32 bits of the flat shared address are the
  // group-segment-relative DS address (aperture base lives in the high word).
  const uint32_t lbase = (uint32_t)(uintptr_t)(&sbuf[0][0][0]);

  auto issue = [&](int t) {
    const uint32_t voff = (uint32_t)((t * D_ + d0) * sizeof(float));
    const uint32_t l0   = lbase + (uint32_t)(
        (((t & (STAGES - 1)) * 4) * D_ + d0) * sizeof(float));
    const uint32_t pl   = (uint32_t)(D_ * sizeof(float));  // plane stride
    asm volatile("global_load_async_to_lds_b64 %0, %1, %2 th:TH_LOAD_LU"
                 :: "v"(l0),          "v"(voff), "s"(pAr) : "memory");
    asm volatile("global_load_async_to_lds_b64 %0, %1, %2 th:TH_LOAD_LU"
                 :: "v"(l0 + pl),     "v"(voff), "s"(pAi) : "memory");
    asm volatile("global_load_async_to_lds_b64 %0, %1, %2 th:TH_LOAD_LU"
                 :: "v"(l0 + 2 * pl), "v"(voff), "s"(pXr) : "memory");
    asm volatile("global_load_async_to_lds_b64 %0, %1, %2 th:TH_LOAD_LU"
                 :: "v"(l0 + 3 * pl), "v"(voff), "s"(pXi) : "memory");
  };

  // Prologue: fill all pipeline stages (8 stages x 4 async ops in flight).
#pragma unroll
  for (int t = 0; t < STAGES; ++t) issue(t);

  const size_t cbase = ((size_t)b * CHUNKS + c) * D_ + d0;
  const f32x2 cy0 = carry[cbase];
  const f32x2 cy1 = carry[cbase + 1];
  float y0r = cy0.x, y0i = cy0.y;
  float y1r = cy1.x, y1i = cy1.y;

  f32x4* Yp = Y2 + (base + d0) / 2;       // f32x4 = 2 complex elements

#pragma unroll
  for (int t = 0; t < LC; ++t) {
    // Async loads complete in order per wave; ensure stage t's 4 loads are
    // done: allow 4*(stages still in flight beyond t) outstanding. With the
    // loop fully unrolled, `rem` is a constant and exactly one s_wait_asynccnt
    // survives per iteration.
    const int rem = LC - 1 - t;
    if (rem >= STAGES - 1) {
      asm volatile("s_wait_asynccnt 28" ::: "memory");
    } else {
      switch (rem) {
        case 6:  asm volatile("s_wait_asynccnt 24" ::: "memory"); break;
        case 5:  asm volatile("s_wait_asynccnt 20" ::: "memory"); break;
        case 4:  asm volatile("s_wait_asynccnt 16" ::: "memory"); break;
        case 3:  asm volatile("s_wait_asynccnt 12" ::: "memory"); break;
        case 2:  asm volatile("s_wait_asynccnt 8"  ::: "memory"); break;
        case 1:  asm volatile("s_wait_asynccnt 4"  ::: "memory"); break;
        default: asm volatile("s_wait_asynccnt 0"  ::: "memory"); break;
      }
    }

    const int s = t & (STAGES - 1);
    const f32x2 va_r = *(const f32x2*)(&sbuf[s][0][d0]);  // ds_load_b64 x4
    const f32x2 va_i = *(const f32x2*)(&sbuf[s][1][d0]);
    const f32x2 vx_r = *(const f32x2*)(&sbuf[s][2][d0]);
    const f32x2 vx_i = *(const f32x2*)(&sbuf[s][3][d0]);

    {
      const float nyr = fmaf(va_r.x, y0r, fmaf(-va_i.x, y0i, vx_r.x));
      const float nyi = fmaf(va_r.x, y0i, fmaf( va_i.x, y0r, vx_i.x));
      y0r = nyr; y0i = nyi;
    }
    {
      const float nyr = fmaf(va_r.y, y1r, fmaf(-va_i.y, y1i, vx_r.y));
      const float nyi = fmaf(va_r.y, y1i, fmaf( va_i.y, y1r, vx_i.y));
      y1r = nyr; y1i = nyi;
    }

    f32x4 o; o.x = y0r; o.y = y0i; o.z = y1r; o.w = y1i;
    __builtin_nontemporal_store(o, Yp + t * D2_);  // b128 NT store

    if (t + STAGES < LC) issue(t + STAGES);        // constant-folds away in tail
  }
}

// ---------------------------------------------------------------------------
// Launch: ws layout = [aggregates 4 MB][carries 2 MB]; fully rewritten each
// call (no cross-call state). All launches on `stream`, graph-capture safe.
// ---------------------------------------------------------------------------
extern "C" void kernel_launch(void* const* d_in, const int* in_sizes, int n_in,
                              void* d_out, int out_size, void* d_ws, size_t ws_size,
                              hipStream_t stream) {
  const float* A_re = (const float*)d_in[0];
  const float* A_im = (const float*)d_in[1];
  const float* X_re = (const float*)d_in[2];
  const float* X_im = (const float*)d_in[3];

  f32x4* agg   = (f32x4*)d_ws;
  f32x2* carry = (f32x2*)((char*)d_ws +
                          (size_t)B_ * CHUNKS * D_ * sizeof(f32x4));
  f32x4* Y2 = (f32x4*)d_out;

  pscan_chunk_agg<<<dim3(B_ * CHUNKS), dim3(128), 0, stream>>>(
      A_re, A_im, X_re, X_im, agg);
  pscan_chunk_carry<<<dim3(B_ * (D_ / 32)), dim3(256), 0, stream>>>(
      agg, carry);
  pscan_apply<<<dim3(B_ * CHUNKS), dim3(128), 0, stream>>>(
      A_re, A_im, X_re, X_im, carry, Y2);
}